// MulByRepeatedAddition_19490561589966
// MI455X (gfx1250) — compile-verified
//
#include <hip/hip_runtime.h>
#include <hip/hip_bf16.h>

// y[r,c] = zeros[c] + n_iter * x[r,c]   (closed form of the 64-step scan)
//
// Memory-bound: 268 MB of traffic -> ~11.5 us floor at 23.3 TB/s.
// Strategy:
//   * one wave32 per 16x16 f32 tile, loaded straight into the WMMA C/D
//     register layout (each global_load_b32 = two dense 64B segments)
//   * non-temporal loads/stores (stream > 192MB L2)
//   * the n_iter scaling rides the (idle) matrix pipe via 4 chained
//     V_WMMA_F32_16X16X4_F32 ops with one-hot selector A matrices

typedef float v2f __attribute__((ext_vector_type(2)));
typedef float v8f __attribute__((ext_vector_type(8)));

__global__ __launch_bounds__(256) void
MulByRepeatedAddition_19490561589966_kernel(const float* __restrict__ x,
                                            const float* __restrict__ zeros,
                                            const int*   __restrict__ n_iter,
                                            float*       __restrict__ y)
{
    constexpr int C = 4096;              // columns
    // tile id: 8 waves per 256-thread block
    const int lane  = threadIdx.x & 31;
    const int tile  = blockIdx.x * 8 + (threadIdx.x >> 5);
    const int rowT  = tile >> 8;         // 4096/16 = 256 column tiles
    const int colT  = tile & 255;

    const int n16   = lane & 15;         // N within tile (B/C/D lane index)
    const int half  = lane >> 4;         // 0: M in [0,8), 1: M in [8,16)
    const int r0    = rowT * 16;
    const int c     = colT * 16 + n16;   // absolute column

    const float nf  = (float)(*n_iter);  // uniform scalar, read on device
    const float zl  = zeros[c];          // 16KB, stays resident in cache

    // ---- load 16x16 tile of x directly in C/D layout -------------------
    // VGPR v, lane L  ->  x[r0 + v + 8*half][c]
    float xr[8];
#pragma unroll
    for (int v = 0; v < 8; ++v) {
        const int row = r0 + v + 8 * half;
        xr[v] = __builtin_nontemporal_load(&x[row * C + c]);
    }

    // ---- accumulator seeded with zeros[c] broadcast over M -------------
    v8f acc = {zl, zl, zl, zl, zl, zl, zl, zl};

    // ---- 4 chained WMMAs: D = sum_p A_p x B_p + C ----------------------
    // B_p = (xr[2p], xr[2p+1]) is a legal 4x16 f32 B matrix holding
    // x-rows {2p, 2p+1, 2p+8, 2p+9}.  A_p is the one-hot selector that
    // routes row K of B_p into D row rowOf(p,K), scaled by n.
    //   A vgpr0 (K=0|2): nonzero at lanes 2p (M=2p) and 24+2p (M=2p+8)
    //   A vgpr1 (K=1|3): nonzero at lanes 2p+1        and 25+2p
#pragma unroll
    for (int p = 0; p < 4; ++p) {
        v2f A, B;
        A.x = (lane == 2 * p     || lane == 24 + 2 * p) ? nf : 0.0f;
        A.y = (lane == 2 * p + 1 || lane == 25 + 2 * p) ? nf : 0.0f;
        B.x = xr[2 * p];
        B.y = xr[2 * p + 1];
        // 8 args: (neg_a, A, neg_b, B, c_mod, C, reuse_a, reuse_b)
        acc = __builtin_amdgcn_wmma_f32_16x16x4_f32(
                  false, A, false, B, (short)0, acc, false, false);
    }

    // ---- store D (same C/D layout -> same coalesced pattern) -----------
#pragma unroll
    for (int v = 0; v < 8; ++v) {
        const int row = r0 + v + 8 * half;
        __builtin_nontemporal_store(acc[v], &y[row * C + c]);
    }
}

extern "C" void kernel_launch(void* const* d_in, const int* in_sizes, int n_in,
                              void* d_out, int out_size, void* d_ws, size_t ws_size,
                              hipStream_t stream) {
    (void)in_sizes; (void)n_in; (void)d_ws; (void)ws_size; (void)out_size;
    const float* x      = (const float*)d_in[0];
    const float* zeros  = (const float*)d_in[1];
    const int*   n_iter = (const int*)d_in[2];
    float*       y      = (float*)d_out;

    // 8192/16 * 4096/16 = 131072 tiles, 8 waves/block -> 16384 blocks
    const int tiles  = (8192 / 16) * (4096 / 16);
    const int blocks = tiles / 8;
    MulByRepeatedAddition_19490561589966_kernel<<<blocks, 256, 0, stream>>>(
        x, zeros, n_iter, y);
}